// MacUnit_21028159881735
// MI455X (gfx1250) — compile-verified
//
#include <hip/hip_runtime.h>
#include <cstdint>

// ---------------- problem constants (match reference hyperparameters) -------
#define NPTS    31          // NUM_POINTS
#define CDIM    1024        // IC * ICF = OC
#define NSTEPS  4
#define BLOCK   256
#define ITERS   4           // grid-stride iterations per thread

#define LOG2E   1.4426950408889634f
#define INV2PI  0.15915494309189535f   // rounds to 0x3e22f983, v_sin/v_cos convention

// ---------------- CDNA5 async global->LDS helpers ---------------------------
// Generic pointers to __shared__ objects carry the LDS byte offset in their
// low 32 bits on amdgcn (addrspacecast AS3->generic truncates), so a 32-bit
// truncation yields a valid DS/async LDS address.
__device__ __forceinline__ uint32_t lds_off(const void* p) {
  return (uint32_t)(uintptr_t)p;
}

__device__ __forceinline__ void async_copy_b128(void* lds_dst, const void* gsrc) {
  asm volatile("global_load_async_to_lds_b128 %0, %1, off"
               :
               : "v"(lds_off(lds_dst)), "v"((uint64_t)(uintptr_t)gsrc)
               : "memory");
}

__device__ __forceinline__ void wait_async_zero() {
  asm volatile("s_wait_asynccnt 0" ::: "memory");
}

// ---------------- kernel -----------------------------------------------------
extern "C" __global__ __launch_bounds__(BLOCK)
void macunit_kernel(const float* __restrict__ data,       // (B, 512)
                    const float* __restrict__ angles,     // (31)
                    const float* __restrict__ velocity,   // (31)
                    const float* __restrict__ attention,  // (1024)
                    const float* __restrict__ alpha,      // (1024)
                    const float* __restrict__ beta,       // (1024)
                    float*       __restrict__ out,        // (B, 1024)
                    int total4)                            // B*512/4
{
  __shared__ __align__(16) float  s_alpha[CDIM];
  __shared__ __align__(16) float  s_beta [CDIM];
  __shared__ __align__(16) float  s_att  [CDIM];
  // {v[i], v[i+1]-v[i], a[i]/2pi, (a[i+1]-a[i])/2pi}
  __shared__ __align__(16) float4 s_lut  [NPTS];

  const int t = threadIdx.x;

  // --- Stage the broadcast tables via async DMA (tracked by ASYNCcnt). ------
  // 256 lanes x b128 == 1024 floats per table, exact cover.
  async_copy_b128(&s_alpha[4 * t], &alpha[4 * t]);
  async_copy_b128(&s_beta [4 * t], &beta [4 * t]);
  async_copy_b128(&s_att  [4 * t], &attention[4 * t]);

  // --- Build segment-interp LUT with wave0 (wave32 shuffle, branch-light). --
  if (t < 32) {
    const int tc = (t < NPTS) ? t : NPTS - 1;   // clamped -> no divergent loads
    float v = velocity[tc];
    float a = angles[tc] * INV2PI;              // pre-scale for v_sin/v_cos
    float v1 = __shfl_down(v, 1);
    float a1 = __shfl_down(a, 1);
    if (t < NPTS - 1) s_lut[t] = make_float4(v, v1 - v, a, a1 - a);
    if (t == 0)       s_lut[NPTS - 1] = make_float4(v, 0.0f, a, 0.0f); // pad
  }

  wait_async_zero();     // our own async transfers
  __syncthreads();       // cross-wave visibility of all LDS staging

  // index == 31.0 saturation wraps to entry 0 in the reference's arithmetic
  const float v_sat = s_lut[0].x;
  const float a_sat = s_lut[0].z;   // already pre-scaled by 1/2pi

  const int stride = gridDim.x * BLOCK;
  int idx = blockIdx.x * BLOCK + t;

#pragma unroll 1
  for (int it = 0; it < ITERS; ++it, idx += stride) {
    if (idx >= total4) break;

    const float4 dv4 = reinterpret_cast<const float4*>(data)[idx];
    if (idx + stride < total4)   // hide HBM latency for the next sweep
      __builtin_prefetch(&reinterpret_cast<const float4*>(data)[idx + stride], 0, 1);

    const int b   = idx >> 7;        // / (512/4)
    const int ic4 = idx & 127;
    const int c0  = ic4 * 8;         // 8 output channels per thread

    const float din[4] = {dv4.x, dv4.y, dv4.z, dv4.w};
    float res[8];

#pragma unroll
    for (int j = 0; j < 4; ++j) {
#pragma unroll
      for (int k = 0; k < 2; ++k) {
        const int   c   = c0 + 2 * j + k;
        // fold sigmoid's -log2(e) scale into the per-channel affine constants:
        // exp(-(a*x+b)) == exp2(fma(alm, x, bem))
        const float alm = s_alpha[c] * (-LOG2E);
        const float bem = s_beta [c] * (-LOG2E);
        float x = din[j];
#pragma unroll
        for (int s = 0; s < NSTEPS; ++s) {
          const float e     = __builtin_amdgcn_exp2f(fmaf(alm, x, bem));
          const float sig   = __builtin_amdgcn_rcpf(1.0f + e);
          const float index = sig * 31.0f;

          // reference boundary arithmetic: end == bgn+1, bgn = min(floor,29)
          float bgn = floorf(index);
          bgn = fminf(bgn, 29.0f);
          const float pos = index - bgn;

          const float4 lut = s_lut[(int)bgn];        // ds_load_b128 gather
          float velo = fmaf(pos, lut.y, lut.x);
          float arg  = fmaf(pos, lut.w, lut.z);      // angle / 2pi
          if (index == 31.0f) { velo = v_sat; arg = a_sat; }  // exact wrap case

          const float ca = __builtin_amdgcn_cosf(arg);
          const float sa = __builtin_amdgcn_sinf(arg);

          // velo*cos + x*velo*sin == velo * fma(x, sin, cos)
          const float stepv = velo * fmaf(x, sa, ca);
          x = fmaf(stepv, 0.25f, x);                 // d += step / NUM_STEPS
        }
        res[2 * j + k] = s_att[c] * x;               // "attention" = diag scale
      }
    }

    float4* op = reinterpret_cast<float4*>(out + (size_t)b * CDIM + c0);
    op[0] = make_float4(res[0], res[1], res[2], res[3]);
    op[1] = make_float4(res[4], res[5], res[6], res[7]);
  }
}

// ---------------- host-side launcher -----------------------------------------
extern "C" void kernel_launch(void* const* d_in, const int* in_sizes, int n_in,
                              void* d_out, int out_size, void* d_ws, size_t ws_size,
                              hipStream_t stream) {
  (void)n_in; (void)out_size; (void)d_ws; (void)ws_size;
  const float* data      = (const float*)d_in[0];
  const float* angles    = (const float*)d_in[1];
  const float* velocity  = (const float*)d_in[2];
  const float* attention = (const float*)d_in[3];
  const float* alpha     = (const float*)d_in[4];
  const float* beta      = (const float*)d_in[5];
  float* out = (float*)d_out;

  const int total4  = in_sizes[0] / 4;                       // B*512/4 float4's
  const int threads = (total4 + ITERS - 1) / ITERS;
  const int blocks  = (threads + BLOCK - 1) / BLOCK;         // 4096 for B=32768

  macunit_kernel<<<blocks, BLOCK, 0, stream>>>(data, angles, velocity, attention,
                                               alpha, beta, out, total4);
}